// GCN_dev_56882546868696
// MI455X (gfx1250) — compile-verified
//
#include <hip/hip_runtime.h>
#include <math.h>

#define N_NODES 100000
#define N_EDGES 1000000
#define EMB 64
#define HID 64
#define LEAKY 0.01f

typedef __attribute__((ext_vector_type(2))) float v2f;
typedef __attribute__((ext_vector_type(8))) float v8f;

// Workspace layout (float offsets). Total ~52.8 MB.
#define OFF_SDEG 0                         // [N_NODES]
#define OFF_RDEG (OFF_SDEG + N_NODES)      // [N_NODES]
#define OFF_H    (OFF_RDEG + N_NODES)      // [N_NODES*64]  (x @ W1) * rsqrt(send_deg)
#define OFF_AGG  (OFF_H + N_NODES * 64)    // [N_NODES*64]  segment-summed rows
#define OFF_Z    (OFF_AGG + N_NODES * 64)  // [N_NODES]     per-node scalar after W2
#define OFF_ACC  (OFF_Z + N_NODES)         // [N_NODES]     final segment sum

__global__ __launch_bounds__(256) void zero_kernel(float* __restrict__ p, int n) {
  int i = blockIdx.x * blockDim.x + threadIdx.x;
  if (i < n) p[i] = 0.0f;
}

__global__ __launch_bounds__(256) void degree_kernel(const int* __restrict__ senders,
                                                     const int* __restrict__ receivers,
                                                     float* __restrict__ sdeg,
                                                     float* __restrict__ rdeg) {
  int e = blockIdx.x * blockDim.x + threadIdx.x;
  if (e < N_EDGES) {
    atomicAdd(&sdeg[senders[e]], 1.0f);
    atomicAdd(&rdeg[receivers[e]], 1.0f);
  }
}

// One wave32 computes a 16x64 tile of h = (embed[node_ids] @ W1) * rsqrt(max(send_deg,1))
// using V_WMMA_F32_16X16X4_F32: 16 K-steps (K=64) x 4 N-tiles.
// W1 is staged in LDS pre-paired along K so each B fragment is one aligned ds_load_b64.
__global__ __launch_bounds__(128) void gemm_wmma_kernel(const int* __restrict__ node_ids,
                                                        const float* __restrict__ embed,
                                                        const float* __restrict__ W1,
                                                        const float* __restrict__ sdeg,
                                                        float* __restrict__ h) {
  // ldsB[kp*64 + c] = { W1[2*kp][c], W1[2*kp+1][c] }, kp = 0..31, c = 0..63  (16 KB)
  __shared__ v2f ldsB[32 * HID];

  int t = threadIdx.x;
  for (int i = t; i < 32 * HID; i += 128) {
    int kp = i >> 6;
    int c = i & 63;
    v2f p;
    p.x = W1[(2 * kp) * HID + c];
    p.y = W1[(2 * kp + 1) * HID + c];
    ldsB[i] = p;
  }
  __syncthreads();

  int wave = t >> 5;
  int lane = t & 31;
  int col = lane & 15;  // N (and M for A-loads) within half-wave
  int hi = lane >> 4;   // 0: K pair {0,1}; 1: K pair {2,3}

  int row0 = (blockIdx.x * 4 + wave) * 16;  // wave-uniform
  if (row0 >= N_NODES) return;              // uniform per wave: EXEC stays all-1s

  // A-fragment source row for this lane (clamped; tail stores are guarded later)
  int arow = row0 + col;
  if (arow >= N_NODES) arow = N_NODES - 1;
  int srow = node_ids[arow];
  const v2f* erow = (const v2f*)(embed + (long)srow * EMB);

  v8f acc0 = {}, acc1 = {}, acc2 = {}, acc3 = {};

#pragma unroll
  for (int kk = 0; kk < 16; ++kk) {
    v2f a = erow[kk * 2 + hi];  // K = kk*4 + hi*2 + {0,1}
    const v2f* brow = &ldsB[(kk * 2 + hi) * HID + col];
    v2f b0 = brow[0];
    v2f b1 = brow[16];
    v2f b2 = brow[32];
    v2f b3 = brow[48];
    acc0 = __builtin_amdgcn_wmma_f32_16x16x4_f32(false, a, false, b0, (short)0, acc0, false, false);
    acc1 = __builtin_amdgcn_wmma_f32_16x16x4_f32(false, a, false, b1, (short)0, acc1, false, false);
    acc2 = __builtin_amdgcn_wmma_f32_16x16x4_f32(false, a, false, b2, (short)0, acc2, false, false);
    acc3 = __builtin_amdgcn_wmma_f32_16x16x4_f32(false, a, false, b3, (short)0, acc3, false, false);
  }

  // Epilogue: row r = row0 + v + 8*hi, col = nt*16 + (lane&15); scale by rsqrt(send_deg)
#pragma unroll
  for (int v = 0; v < 8; ++v) {
    int r = row0 + v + hi * 8;
    int rc = r < N_NODES ? r : N_NODES - 1;
    float rs = rsqrtf(fmaxf(sdeg[rc], 1.0f));
    if (r < N_NODES) {
      float* out = h + (long)r * HID + col;
      out[0] = acc0[v] * rs;
      out[16] = acc1[v] * rs;
      out[32] = acc2[v] * rs;
      out[48] = acc3[v] * rs;
    }
  }
}

// 16 threads per edge; each moves a float4 chunk of the 64-float row via f32 atomics.
__global__ __launch_bounds__(256) void scatter_rows_kernel(const int* __restrict__ senders,
                                                           const int* __restrict__ receivers,
                                                           const float* __restrict__ h,
                                                           float* __restrict__ agg) {
  long idx = (long)blockIdx.x * blockDim.x + threadIdx.x;
  int e = (int)(idx >> 4);
  int c = (int)(idx & 15);
  if (e < N_EDGES) {
    int s = senders[e];
    int r = receivers[e];
    float4 v = *(const float4*)(h + (long)s * HID + c * 4);
    float* dst = agg + (long)r * HID + c * 4;
    atomicAdd(dst + 0, v.x);
    atomicAdd(dst + 1, v.y);
    atomicAdd(dst + 2, v.z);
    atomicAdd(dst + 3, v.w);
  }
}

// One wave32 per node: recv-scale, leaky_relu, dot with W2, + b2.
__global__ __launch_bounds__(256) void node_head_kernel(const float* __restrict__ agg,
                                                        const float* __restrict__ rdeg,
                                                        const float* __restrict__ W2,
                                                        const float* __restrict__ b2,
                                                        float* __restrict__ z) {
  long gid = (long)blockIdx.x * blockDim.x + threadIdx.x;
  int n = (int)(gid >> 5);
  int lane = (int)(gid & 31);
  if (n >= N_NODES) return;
  float rs = rsqrtf(fmaxf(rdeg[n], 1.0f));
  float sum = 0.0f;
#pragma unroll
  for (int d = lane; d < HID; d += 32) {
    float x = agg[(long)n * HID + d] * rs;
    x = x > 0.0f ? x : LEAKY * x;
    sum += x * W2[d];
  }
#pragma unroll
  for (int off = 16; off > 0; off >>= 1) sum += __shfl_xor(sum, off, 32);
  if (lane == 0) z[n] = sum + b2[0];
}

__global__ __launch_bounds__(256) void scatter_scalar_kernel(const int* __restrict__ senders,
                                                             const int* __restrict__ receivers,
                                                             const float* __restrict__ z,
                                                             float* __restrict__ acc) {
  int e = blockIdx.x * blockDim.x + threadIdx.x;
  if (e < N_EDGES) atomicAdd(&acc[receivers[e]], z[senders[e]]);
}

__global__ __launch_bounds__(256) void sigmoid_kernel(const float* __restrict__ acc,
                                                      float* __restrict__ out) {
  int n = blockIdx.x * blockDim.x + threadIdx.x;
  if (n < N_NODES) out[n] = 1.0f / (1.0f + expf(-acc[n]));
}

extern "C" void kernel_launch(void* const* d_in, const int* in_sizes, int n_in,
                              void* d_out, int out_size, void* d_ws, size_t ws_size,
                              hipStream_t stream) {
  (void)in_sizes; (void)n_in; (void)out_size; (void)ws_size;

  const int* node_ids = (const int*)d_in[0];
  const int* senders = (const int*)d_in[1];
  const int* receivers = (const int*)d_in[2];
  const float* embed = (const float*)d_in[3];
  const float* W1 = (const float*)d_in[4];
  const float* W2 = (const float*)d_in[5];
  const float* b2 = (const float*)d_in[6];
  float* out = (float*)d_out;

  float* ws = (float*)d_ws;
  float* sdeg = ws + OFF_SDEG;
  float* rdeg = ws + OFF_RDEG;
  float* h = ws + OFF_H;
  float* agg = ws + OFF_AGG;
  float* z = ws + OFF_Z;
  float* acc = ws + OFF_ACC;

  // Zero the accumulated regions every launch (harness does not re-poison/restore ws).
  zero_kernel<<<(2 * N_NODES + 255) / 256, 256, 0, stream>>>(sdeg, 2 * N_NODES);  // sdeg+rdeg contiguous
  zero_kernel<<<(N_NODES * 64 + 255) / 256, 256, 0, stream>>>(agg, N_NODES * 64);
  zero_kernel<<<(N_NODES + 255) / 256, 256, 0, stream>>>(acc, N_NODES);

  degree_kernel<<<(N_EDGES + 255) / 256, 256, 0, stream>>>(senders, receivers, sdeg, rdeg);

  // 16 rows per wave, 4 waves per block -> 64 rows/block
  int gemm_blocks = (N_NODES + 63) / 64;
  gemm_wmma_kernel<<<gemm_blocks, 128, 0, stream>>>(node_ids, embed, W1, sdeg, h);

  long sc_threads = (long)N_EDGES * 16;
  scatter_rows_kernel<<<(int)((sc_threads + 255) / 256), 256, 0, stream>>>(senders, receivers, h, agg);

  long nh_threads = (long)N_NODES * 32;
  node_head_kernel<<<(int)((nh_threads + 255) / 256), 256, 0, stream>>>(agg, rdeg, W2, b2, z);

  scatter_scalar_kernel<<<(N_EDGES + 255) / 256, 256, 0, stream>>>(senders, receivers, z, acc);

  sigmoid_kernel<<<(N_NODES + 255) / 256, 256, 0, stream>>>(acc, out);
}